// ClassCaps_38646115729905
// MI455X (gfx1250) — compile-verified
//
#include <hip/hip_runtime.h>
#include <hip/hip_bf16.h>
#include <math.h>

// Problem constants (from reference)
#define B_   64
#define J_   2048
#define DIN  16
#define C_   32
#define DO_  32
#define EPS_ 1e-7f

typedef float v2f __attribute__((ext_vector_type(2)));
typedef float v8f __attribute__((ext_vector_type(8)));

// ---------------- K1: u_hat GEMM (WMMA f32 16x16x4) + iter-0 s partials -----
// wave handles one (c, j) batch GEMM 64x32x16 per j, NJW j's sequentially.
#define NJW     4
#define WPB     8                    // waves per block (256 threads)
#define JCHUNK1 (NJW * WPB)          // 32 j per block
#define NJC1    (J_ / JCHUNK1)       // 64 chunks

__global__ __launch_bounds__(256)
void caps_uhat_gemm(const float* __restrict__ x, const float* __restrict__ W,
                    float* __restrict__ U, float* __restrict__ P1) {
    const int c    = blockIdx.x % C_;
    const int jc   = blockIdx.x / C_;
    const int wave = threadIdx.x >> 5;
    const int lane = threadIdx.x & 31;
    const int l16  = lane & 15;
    const int lh   = lane >> 4;                 // 0 or 1
    const int jbase = jc * JCHUNK1 + wave * NJW;

    v8f sacc[4][2];
    #pragma unroll
    for (int m = 0; m < 4; ++m)
        #pragma unroll
        for (int n = 0; n < 2; ++n) sacc[m][n] = {};

    for (int jj = 0; jj < NJW; ++jj) {
        const int j = jbase + jj;
        // ---- A fragments: 16x4 f32 (M=b tile, K=i). VGPR0: K=4k+2*lh, VGPR1: +1
        v2f a[4][4];
        #pragma unroll
        for (int m = 0; m < 4; ++m)
            #pragma unroll
            for (int k = 0; k < 4; ++k) {
                const int bb = 16 * m + l16;
                const int i  = 4 * k + (lh << 1);
                const float* p = x + ((size_t)bb * J_ + j) * DIN + i;
                a[m][k].x = p[0];
                a[m][k].y = p[1];
            }
        // ---- B fragments: 4x16 f32 (K=i, N=o), mirrored layout
        const float* wj = W + ((size_t)c * J_ + j) * (DO_ * DIN);
        v2f bf[2][4];
        #pragma unroll
        for (int n = 0; n < 2; ++n)
            #pragma unroll
            for (int k = 0; k < 4; ++k) {
                const int o = 16 * n + l16;
                const int i = 4 * k + (lh << 1);
                const float* p = wj + (size_t)o * DIN + i;
                bf[n][k].x = p[0];
                bf[n][k].y = p[1];
            }
        if (jj + 1 < NJW)
            __builtin_prefetch(wj + (size_t)DO_ * DIN, 0, 1);   // global_prefetch next W slab

        // ---- 4x2 accumulator tiles, 4 k-steps each
        #pragma unroll
        for (int m = 0; m < 4; ++m)
            #pragma unroll
            for (int n = 0; n < 2; ++n) {
                v8f t = {};
                #pragma unroll
                for (int k = 0; k < 4; ++k)
                    t = __builtin_amdgcn_wmma_f32_16x16x4_f32(
                            false, a[m][k], false, bf[n][k], (short)0, t, false, false);
                sacc[m][n] += t;
                // store D: row M = r + 8*lh (+16m), col N = l16 (+16n)
                #pragma unroll
                for (int r = 0; r < 8; ++r) {
                    const int bb = 16 * m + r + lh * 8;
                    const int o  = 16 * n + l16;
                    U[(((size_t)bb * J_ + j) * C_ + c) * DO_ + o] = t[r];
                }
            }
    }

    // iter-0: c = softmax(0) = 1/C exactly -> s0 partial = sum_j u_hat / C
    __shared__ float red[WPB][B_ * DO_];        // 64 KB
    #pragma unroll
    for (int m = 0; m < 4; ++m)
        #pragma unroll
        for (int n = 0; n < 2; ++n)
            #pragma unroll
            for (int r = 0; r < 8; ++r) {
                const int bb = 16 * m + r + lh * 8;
                const int o  = 16 * n + l16;
                red[wave][bb * DO_ + o] = sacc[m][n][r] * (1.0f / (float)C_);
            }
    __syncthreads();
    // deterministic fixed-order cross-wave sum -> per-block partial
    for (int e = threadIdx.x; e < B_ * DO_; e += 256) {
        float s = 0.f;
        #pragma unroll
        for (int w = 0; w < WPB; ++w) s += red[w][e];
        P1[((size_t)c * NJC1 + jc) * (B_ * DO_) + e] = s;   // P1[c][jc][b][o]
    }
}

// ---------------- K2: deterministic chunk-reduce + squash ------------------
// one wave per (b,c); lane = o. addr = b*sb + c*sc + ch*sch + o
__global__ __launch_bounds__(256)
void caps_reduce_squash(const float* __restrict__ P, float* __restrict__ v,
                        float* __restrict__ out,
                        int nch, int sch, int sb, int sc) {
    const int gw   = (blockIdx.x * 256 + threadIdx.x) >> 5;
    const int lane = threadIdx.x & 31;
    const int b = gw / C_;
    const int c = gw % C_;
    const float* p = P + (size_t)b * sb + (size_t)c * sc + lane;
    float s = 0.f;
    for (int ch = 0; ch < nch; ++ch) s += p[(size_t)ch * sch];
    // squash: scale = s2/(1+s2)/sqrt(s2+eps), s2 = sum_o s^2
    float s2 = s * s;
    #pragma unroll
    for (int m = 16; m >= 1; m >>= 1) s2 += __shfl_xor(s2, m, 32);
    const float scale = s2 / ((1.0f + s2) * sqrtf(s2 + EPS_));
    const float val = scale * s;
    v[((size_t)b * C_ + c) * DO_ + lane] = val;
    if (out) out[((size_t)b * C_ + c) * DO_ + lane] = val;
}

// ---------------- K3: fused logit-update + softmax(C) + s accumulation -----
// wave per j (lane = o); block = (b, j-chunk of 128), 8 waves x 16 j.
#define NJC2    16
#define JCHUNK2 (J_ / NJC2)          // 128
#define NJW2    (JCHUNK2 / WPB)      // 16

__global__ __launch_bounds__(256)
void caps_route(const float* __restrict__ U, const float* __restrict__ v,
                float* __restrict__ logits, float* __restrict__ P2,
                int read_logits, int write_logits) {
    const int b    = blockIdx.x / NJC2;
    const int jc   = blockIdx.x % NJC2;
    const int wave = threadIdx.x >> 5;
    const int lane = threadIdx.x & 31;          // = o

    float vv[C_];
    #pragma unroll
    for (int c = 0; c < C_; ++c) vv[c] = v[((size_t)b * C_ + c) * DO_ + lane];
    float sacc[C_];
    #pragma unroll
    for (int c = 0; c < C_; ++c) sacc[c] = 0.f;

    for (int jj = 0; jj < NJW2; ++jj) {
        const int j = jc * JCHUNK2 + wave * NJW2 + jj;
        const float* uj = U + ((size_t)b * J_ + j) * (C_ * DO_);
        if (jj + 1 < NJW2) __builtin_prefetch(uj + C_ * DO_, 0, 1);

        float u[C_];
        #pragma unroll
        for (int c = 0; c < C_; ++c) u[c] = uj[c * DO_ + lane];

        // logit[c] = b_old[c] + sum_o v[c][o]*u[c][o] (butterfly over 32 lanes)
        float lg[C_];
        #pragma unroll
        for (int c = 0; c < C_; ++c) {
            float p = vv[c] * u[c];
            #pragma unroll
            for (int m = 16; m >= 1; m >>= 1) p += __shfl_xor(p, m, 32);
            lg[c] = p;
        }
        if (read_logits) {
            const float* lp = logits + ((size_t)b * J_ + j) * C_;
            #pragma unroll
            for (int c = 0; c < C_; ++c) lg[c] += lp[c];
        }
        if (write_logits) {
            float lgmine = 0.f;
            #pragma unroll
            for (int c = 0; c < C_; ++c) lgmine = (lane == c) ? lg[c] : lgmine;
            logits[((size_t)b * J_ + j) * C_ + lane] = lgmine;   // coalesced 128B
        }
        // softmax over c (redundant per lane, all in registers)
        float mx = lg[0];
        #pragma unroll
        for (int c = 1; c < C_; ++c) mx = fmaxf(mx, lg[c]);
        float e[C_], den = 0.f;
        #pragma unroll
        for (int c = 0; c < C_; ++c) { e[c] = __expf(lg[c] - mx); den += e[c]; }
        const float inv = 1.0f / den;
        #pragma unroll
        for (int c = 0; c < C_; ++c) sacc[c] = fmaf(e[c] * inv, u[c], sacc[c]);
    }

    __shared__ float red[WPB][C_ * DO_];        // 32 KB
    #pragma unroll
    for (int c = 0; c < C_; ++c) red[wave][c * DO_ + lane] = sacc[c];
    __syncthreads();
    for (int t = threadIdx.x; t < C_ * DO_; t += 256) {
        float s = 0.f;
        #pragma unroll
        for (int w = 0; w < WPB; ++w) s += red[w][t];
        P2[((size_t)b * NJC2 + jc) * (C_ * DO_) + t] = s;   // P2[b][jc][c][o]
    }
}

extern "C" void kernel_launch(void* const* d_in, const int* in_sizes, int n_in,
                              void* d_out, int out_size, void* d_ws, size_t ws_size,
                              hipStream_t stream) {
    (void)in_sizes; (void)n_in; (void)out_size; (void)ws_size;
    const float* x = (const float*)d_in[0];   // [B, J, DIN]
    const float* W = (const float*)d_in[1];   // [C, J, DO, DIN]
    float* out = (float*)d_out;               // [B, C, DO]

    char* ws = (char*)d_ws;
    size_t off = 0;
    float* U  = (float*)(ws + off); off += (size_t)B_ * J_ * C_ * DO_ * 4;   // 512 MB  u_hat[b][j][c][o]
    float* P1 = (float*)(ws + off); off += (size_t)C_ * NJC1 * B_ * DO_ * 4; // 16 MB
    float* P2 = (float*)(ws + off); off += (size_t)B_ * NJC2 * C_ * DO_ * 4; // 4 MB
    float* Lg = (float*)(ws + off); off += (size_t)B_ * J_ * C_ * 4;         // 16 MB   logits[b][j][c]
    float* v  = (float*)(ws + off); off += (size_t)B_ * C_ * DO_ * 4;        // 256 KB

    dim3 blk(256);
    // iter 0: GEMM + uniform-c partial s0
    caps_uhat_gemm<<<dim3(C_ * NJC1), blk, 0, stream>>>(x, W, U, P1);
    // v0 = squash(reduce(P1));  P1[c][jc][b][o]: sb=DO, sc=NJC1*B*DO, sch=B*DO
    caps_reduce_squash<<<dim3((B_ * C_) / WPB), blk, 0, stream>>>(
        P1, v, nullptr, NJC1, B_ * DO_, DO_, NJC1 * B_ * DO_);
    // iter 1: b1 = v0.u_hat (write logits), s1 partials
    caps_route<<<dim3(B_ * NJC2), blk, 0, stream>>>(U, v, Lg, P2, 0, 1);
    // v1 = squash(reduce(P2));  P2[b][jc][c][o]: sb=NJC2*C*DO, sc=DO, sch=C*DO
    caps_reduce_squash<<<dim3((B_ * C_) / WPB), blk, 0, stream>>>(
        P2, v, nullptr, NJC2, C_ * DO_, NJC2 * C_ * DO_, DO_);
    // iter 2: b2 = b1 + v1.u_hat (read logits), s2 partials
    caps_route<<<dim3(B_ * NJC2), blk, 0, stream>>>(U, v, Lg, P2, 1, 0);
    // v2 = squash(reduce(P2)) -> output
    caps_reduce_squash<<<dim3((B_ * C_) / WPB), blk, 0, stream>>>(
        P2, v, out, NJC2, C_ * DO_, NJC2 * C_ * DO_, DO_);
}